// GraphConvLayer_7275674599909
// MI455X (gfx1250) — compile-verified
//
#include <hip/hip_runtime.h>

typedef __attribute__((ext_vector_type(2))) float v2f;
typedef __attribute__((ext_vector_type(8))) float v8f;

#define NNODES   100000
#define NEDGES   1250000
#define IN_DIM   64
#define OUT_DIM  64
#define EDGE_DIM 32

// ---------------------------------------------------------------------------
// Kernel 1: zero the aggregation buffer (d_out reused as agg) and the degree
// buffer (in d_ws). Must run every call: harness poisons buffers and we
// accumulate with atomics.
// ---------------------------------------------------------------------------
__global__ void __launch_bounds__(256) gc_zero_kernel(float* __restrict__ agg,
                                                      float* __restrict__ deg) {
  const long long total = (long long)NNODES * OUT_DIM;
  const long long stride = (long long)gridDim.x * blockDim.x;
  for (long long i = (long long)blockIdx.x * blockDim.x + threadIdx.x; i < total; i += stride)
    agg[i] = 0.0f;
  for (long long i = (long long)blockIdx.x * blockDim.x + threadIdx.x; i < NNODES; i += stride)
    deg[i] = 0.0f;
}

// ---------------------------------------------------------------------------
// Kernel 2: per-edge. Each wave32 handles 16 edges:
//   msg[16x64] = edge_attr[16x32] @ edge_w.T[32x64]   (f32 WMMA, K-split by 4)
//   val = msg + edge_b + x[row]
//   atomicAdd(agg[col], val); atomicAdd(deg[col], 1)
// A-matrix 16x4 f32 layout: lanes 0-15 hold K0(K1 in v1), lanes 16-31 K2/K3.
// B-matrix 4x16 f32 layout: vgpr0 = rows K0(lo)/K2(hi), vgpr1 = K1/K3.
// C/D 16x16 f32 layout: vgpr v, lane L -> row v + 8*(L>>4), col L&15.
// ---------------------------------------------------------------------------
__global__ void __launch_bounds__(256) gc_edge_kernel(
    const float* __restrict__ x,          // [N,64]
    const long long* __restrict__ ei,     // [2,E]
    const float* __restrict__ edge_attr,  // [E,32]
    const float* __restrict__ edge_w,     // [64,32]  (out,in)
    const float* __restrict__ edge_b,     // [64]
    float* __restrict__ agg,              // [N,64]  accumulator (d_out)
    float* __restrict__ deg)              // [N]
{
  const int lane = threadIdx.x & 31;
  const int wave = threadIdx.x >> 5;
  const long long wid = (long long)blockIdx.x * 8 + wave;
  const long long num_waves = NEDGES / 16;  // 78125, exact
  if (wid >= num_waves) return;             // wave-uniform: EXEC all-1s below
  const long long e0 = wid * 16;

  const int m  = lane & 15;   // matrix row (A) / column (B,C)
  const int kh = lane >> 4;   // K-half select

  // --- load A tiles: edge_attr row (e0+m), 8 tiles of K=4 ---
  v2f a[8];
  const float* arow = edge_attr + (e0 + m) * EDGE_DIM;
  #pragma unroll
  for (int kt = 0; kt < 8; ++kt)
    a[kt] = *(const v2f*)(arow + kt * 4 + kh * 2);

  // --- GEMM: 4 output tiles of 16 columns each ---
  v8f c[4];
  #pragma unroll
  for (int nt = 0; nt < 4; ++nt) {
    v8f acc = {};
    const float* wrow = edge_w + (nt * 16 + m) * EDGE_DIM;  // B[k][n]=edge_w[n][k]
    #pragma unroll
    for (int kt = 0; kt < 8; ++kt) {
      v2f b = *(const v2f*)(wrow + kt * 4 + kh * 2);
      acc = __builtin_amdgcn_wmma_f32_16x16x4_f32(
          /*neg_a=*/false, a[kt], /*neg_b=*/false, b,
          /*c_mod=*/(short)0, acc, /*reuse_a=*/false, /*reuse_b=*/false);
    }
    c[nt] = acc;
  }

  // --- add edge bias + gathered source features, scatter-add to destination ---
  const long long* rowp = ei;           // edge_index[0]
  const long long* colp = ei + NEDGES;  // edge_index[1]
  #pragma unroll
  for (int v = 0; v < 8; ++v) {
    const long long e = e0 + v + 8 * kh;  // edge handled by this (vgpr, lane-half)
    const long long r = rowp[e];
    const long long cdst = colp[e];
    const float* xrow = x + r * IN_DIM;
    float* aggrow = agg + cdst * OUT_DIM;
    #pragma unroll
    for (int nt = 0; nt < 4; ++nt) {
      const int n = nt * 16 + m;
      const float val = c[nt][v] + edge_b[n] + xrow[n];
      atomicAdd(&aggrow[n], val);
    }
  }

  // --- degree: one atomic per edge (lanes 0-15 own one edge each) ---
  if (lane < 16) {
    atomicAdd(&deg[colp[e0 + lane]], 1.0f);
  }
}

// ---------------------------------------------------------------------------
// Kernel 3: per-node. Each wave handles 16 nodes:
//   out = (agg / max(deg,1)) @ lin_w.T + lin_b      (in-place on d_out)
// Reads complete before stores: stores depend on WMMA accumulators which
// depend on the loads.
// ---------------------------------------------------------------------------
__global__ void __launch_bounds__(256) gc_node_kernel(
    const float* __restrict__ lin_w,  // [64,64] (out,in)
    const float* __restrict__ lin_b,  // [64]
    const float* __restrict__ deg,    // [N]
    float* __restrict__ out)          // [N,64]: agg in, result out
{
  const int lane = threadIdx.x & 31;
  const int wave = threadIdx.x >> 5;
  const long long wid = (long long)blockIdx.x * 8 + wave;
  const long long num_waves = NNODES / 16;  // 6250, exact
  if (wid >= num_waves) return;
  const long long n0 = wid * 16;  // first node of this wave's tile

  const int m  = lane & 15;
  const int kh = lane >> 4;

  const float d = deg[n0 + m];
  const float invd = 1.0f / fmaxf(d, 1.0f);

  // --- load + normalize A tiles: agg row (n0+m), 16 tiles of K=4 ---
  v2f a[16];
  const float* arow = out + (n0 + m) * OUT_DIM;
  #pragma unroll
  for (int kt = 0; kt < 16; ++kt) {
    v2f t = *(const v2f*)(arow + kt * 4 + kh * 2);
    a[kt] = t * invd;
  }

  v8f c[4];
  #pragma unroll
  for (int nt = 0; nt < 4; ++nt) {
    v8f acc = {};
    const float* wrow = lin_w + (nt * 16 + m) * IN_DIM;  // B[k][n]=lin_w[n][k]
    #pragma unroll
    for (int kt = 0; kt < 16; ++kt) {
      v2f b = *(const v2f*)(wrow + kt * 4 + kh * 2);
      acc = __builtin_amdgcn_wmma_f32_16x16x4_f32(
          false, a[kt], false, b, (short)0, acc, false, false);
    }
    c[nt] = acc;
  }

  // --- add bias, store (overwrites agg rows of this wave only) ---
  #pragma unroll
  for (int v = 0; v < 8; ++v) {
    float* orow = out + (n0 + v + 8 * kh) * OUT_DIM;
    #pragma unroll
    for (int nt = 0; nt < 4; ++nt) {
      const int n = nt * 16 + m;
      orow[n] = c[nt][v] + lin_b[n];
    }
  }
}

// ---------------------------------------------------------------------------
extern "C" void kernel_launch(void* const* d_in, const int* in_sizes, int n_in,
                              void* d_out, int out_size, void* d_ws, size_t ws_size,
                              hipStream_t stream) {
  const float*     x         = (const float*)d_in[0];
  const long long* edge_idx  = (const long long*)d_in[1];
  const float*     edge_attr = (const float*)d_in[2];
  const float*     lin_w     = (const float*)d_in[3];
  const float*     lin_b     = (const float*)d_in[4];
  const float*     edge_w    = (const float*)d_in[5];
  const float*     edge_b    = (const float*)d_in[6];

  float* agg = (float*)d_out;   // reuse output buffer as the accumulator
  float* deg = (float*)d_ws;    // N floats of scratch

  // 1) zero accumulators
  gc_zero_kernel<<<4096, 256, 0, stream>>>(agg, deg);

  // 2) edge phase: 78125 waves, 8 waves (256 threads) per block
  const int edge_blocks = (NEDGES / 16 + 7) / 8;  // 9766
  gc_edge_kernel<<<edge_blocks, 256, 0, stream>>>(x, edge_idx, edge_attr,
                                                  edge_w, edge_b, agg, deg);

  // 3) node phase: 6250 waves
  const int node_blocks = (NNODES / 16 + 7) / 8;  // 782
  gc_node_kernel<<<node_blocks, 256, 0, stream>>>(lin_w, lin_b, deg, agg);
}